// EQGraphNet_22832046145598
// MI455X (gfx1250) — compile-verified
//
#include <hip/hip_runtime.h>
#include <hip/hip_bf16.h>

// ---------------------------------------------------------------------------
// EQGraphNet on MI455X (gfx1250), f32 WMMA path, templated layers +
// fragment-major weight repacking.
//
//   repack_*   : rewrite layer weights into WMMA B-fragment lane order, so
//                the GEMM hot loop does ONE coalesced v2f load per fragment.
//   conv_wmma  : Y[M,Cout] = act(rows(in)) @ W' + b      (M = 512*Nout)
//   gcn_wmma   : h[M,C] = dinv_i * sum_{s=-1,0,1} WMMA(scaled-shifted A, Wg)
//                + hi[M,C] + gnn_b   (tridiagonal GCN as 3 shifted WMMAs)
//
// Bandwidth-bound workload (AI ~ 8 FLOP/B, 23.3 TB/s HBM) => exact-f32
// V_WMMA_F32_16X16X4_F32 is the right precision; this round only cleans the
// instruction mix around the WMMAs (fewer vmem + valu per matrix op).
// ---------------------------------------------------------------------------

typedef float v2f __attribute__((ext_vector_type(2)));
typedef float v8f __attribute__((ext_vector_type(8)));

template <int N>
__device__ __forceinline__ float chain_dinv(int n) {
    // deg(i) = 1 (self loop) + 0.5 per existing chain neighbor
    float deg = 1.0f + 0.5f * ((n > 0 ? 1.0f : 0.0f) + (n < N - 1 ? 1.0f : 0.0f));
    return rsqrtf(deg);
}

// ---------------------------------------------------------------------------
// Weight repack kernels: frag[((ct*KT + kt)*32 + lane)] = v2f{B[k][col], B[k+1][col]}
// with k = kt*4 + 2*(lane>>4), col = ct*16 + (lane&15). Zero-padded past K.
// ---------------------------------------------------------------------------
template <int CIN, int COUT>
__global__ void repack_conv_w(const float* __restrict__ w,   // [COUT, CIN, 2]
                              float* __restrict__ frag) {
    constexpr int K  = 2 * CIN;
    constexpr int KT = (K + 3) / 4;
    constexpr int CT = COUT / 16;
    const int t = blockIdx.x * blockDim.x + threadIdx.x;
    if (t >= CT * KT * 32) return;
    const int lane = t & 31;
    const int kt   = (t >> 5) % KT;
    const int ct   = (t >> 5) / KT;
    const int half = lane >> 4, idx = lane & 15;
    const int k    = kt * 4 + 2 * half;
    const int col  = ct * 16 + idx;
    // B[k][col] = w[col][k % CIN][k / CIN]
    float b0 = (k     < K) ? w[(col * CIN + (k % CIN)) * 2 + (k / CIN)]             : 0.f;
    float b1 = (k + 1 < K) ? w[(col * CIN + ((k + 1) % CIN)) * 2 + ((k + 1) / CIN)] : 0.f;
    frag[2 * t]     = b0;
    frag[2 * t + 1] = b1;
}

template <int C>
__global__ void repack_gcn_w(const float* __restrict__ gw,   // [C, C] (f, o)
                             float* __restrict__ frag) {
    constexpr int KT = C / 4;
    constexpr int CT = C / 16;
    const int t = blockIdx.x * blockDim.x + threadIdx.x;
    if (t >= CT * KT * 32) return;
    const int lane = t & 31;
    const int kt   = (t >> 5) % KT;
    const int ct   = (t >> 5) / KT;
    const int half = lane >> 4, idx = lane & 15;
    const int k    = kt * 4 + 2 * half;
    const int col  = ct * 16 + idx;
    frag[2 * t]     = gw[k * C + col];
    frag[2 * t + 1] = gw[(k + 1) * C + col];
}

// ---------------------------------------------------------------------------
// Conv1d(k=2, stride=2) as GEMM. 4 waves/block; each wave owns a 16x16 tile.
// FIRST=true: input is x in [B, CIN, L]. Otherwise node-major [B, NIN, CIN]:
// output row m=b*NOUT+n reads the contiguous 2*CIN floats at node 2n.
// ---------------------------------------------------------------------------
template <int NOUT, int NIN, int CIN, int COUT, bool RELU, bool FIRST, int L>
__global__ void conv_wmma(const float* __restrict__ in,
                          const float* __restrict__ wfrag,  // fragment-major
                          const float* __restrict__ bias,   // [COUT]
                          float* __restrict__ out) {        // [M, COUT]
    constexpr int K  = 2 * CIN;
    constexpr int KT = (K + 3) / 4;
    const int lane = threadIdx.x & 31;
    const int wave = threadIdx.x >> 5;
    const int half = lane >> 4;
    const int idx  = lane & 15;
    const int m0   = (blockIdx.x * 4 + wave) * 16;
    const int col0 = blockIdx.y * 16;

    const int mA  = m0 + idx;         // A-matrix row owned by this lane
    const int b   = mA / NOUT;        // compile-time NOUT -> mul/shift
    const int n   = mA % NOUT;
    const int col = col0 + idx;       // B/C/D column owned by this lane

    const float* rowp = FIRST ? in : (in + ((long)b * NIN + 2 * n) * CIN);
    const v2f*   wt   = (const v2f*)wfrag + ((long)(col0 >> 4) * KT) * 32 + lane;

    v8f acc0 = {0.f, 0.f, 0.f, 0.f, 0.f, 0.f, 0.f, 0.f};
    v8f acc1 = {0.f, 0.f, 0.f, 0.f, 0.f, 0.f, 0.f, 0.f};

#pragma unroll
    for (int kt = 0; kt < KT; ++kt) {
        const int k = kt * 4 + 2 * half;  // this lane-half supplies K = k, k+1
        v2f a;
        if (FIRST) {
            // A[m][k]: ci = k % CIN, t = k / CIN ; x is [B, CIN, L]
            float a0 = 0.f, a1 = 0.f;
            if (k < K)
                a0 = in[((long)b * CIN + (k % CIN)) * L + 2 * n + (k / CIN)];
            if (k + 1 < K)
                a1 = in[((long)b * CIN + ((k + 1) % CIN)) * L + 2 * n + ((k + 1) / CIN)];
            a.x = a0; a.y = a1;
        } else {
            a = *(const v2f*)(rowp + k);      // 8B-aligned: CIN%16==0, k even
            if (RELU) { a.x = fmaxf(a.x, 0.f); a.y = fmaxf(a.y, 0.f); }
        }
        const v2f bm = wt[kt * 32];           // one coalesced fragment load

        if (kt & 1)
            acc1 = __builtin_amdgcn_wmma_f32_16x16x4_f32(
                     false, a, false, bm, (short)0, acc1, false, false);
        else
            acc0 = __builtin_amdgcn_wmma_f32_16x16x4_f32(
                     false, a, false, bm, (short)0, acc0, false, false);
    }

    const float bv = bias[col];
#pragma unroll
    for (int v = 0; v < 8; ++v) {
        const int row = m0 + v + 8 * half;    // C/D layout: M = v + 8*half
        out[(long)row * COUT + col] = acc0[v] + acc1[v] + bv;
    }
}

// ---------------------------------------------------------------------------
// Fused GCN: out = D^{-1/2} (A+I)_w D^{-1/2} (hi Wg) + hi + gnn_b
// Chain graph => tridiagonal mix, as 3 independent WMMA accumulators over
// row-shifted, per-row dinv-scaled A fragments.
// ---------------------------------------------------------------------------
template <int N, int C>
__global__ void gcn_wmma(const float* __restrict__ hi,     // [M, C]
                         const float* __restrict__ wfrag,  // fragment-major Wg
                         const float* __restrict__ gb,     // [C]
                         float* __restrict__ out) {        // [M, C]
    constexpr int KT = C / 4;
    const int lane = threadIdx.x & 31;
    const int wave = threadIdx.x >> 5;
    const int half = lane >> 4;
    const int idx  = lane & 15;
    const int m0   = (blockIdx.x * 4 + wave) * 16;
    const int col0 = blockIdx.y * 16;

    const int mA  = m0 + idx;
    const int n   = mA % N;                   // compile-time N -> mul/shift
    const int col = col0 + idx;

    const bool hasL = (n > 0);
    const bool hasR = (n < N - 1);
    const float sc_s = chain_dinv<N>(n);                         // dinv[j], j=i
    const float sc_l = hasL ? 0.5f * chain_dinv<N>(n - 1) : 0.f;
    const float sc_r = hasR ? 0.5f * chain_dinv<N>(n + 1) : 0.f;

    const float* rowS = hi + (long)mA * C;
    const float* rowL = hasL ? (rowS - C) : rowS;  // contribution zeroed by sc_l
    const float* rowR = hasR ? (rowS + C) : rowS;
    const v2f*   wt   = (const v2f*)wfrag + ((long)(col0 >> 4) * KT) * 32 + lane;

    v8f accS = {0.f, 0.f, 0.f, 0.f, 0.f, 0.f, 0.f, 0.f};
    v8f accL = {0.f, 0.f, 0.f, 0.f, 0.f, 0.f, 0.f, 0.f};
    v8f accR = {0.f, 0.f, 0.f, 0.f, 0.f, 0.f, 0.f, 0.f};

#pragma unroll
    for (int kt = 0; kt < KT; ++kt) {
        const int k = kt * 4 + 2 * half;
        v2f aS = *(const v2f*)(rowS + k);
        v2f aL = *(const v2f*)(rowL + k);
        v2f aR = *(const v2f*)(rowR + k);
        aS.x *= sc_s; aS.y *= sc_s;
        aL.x *= sc_l; aL.y *= sc_l;
        aR.x *= sc_r; aR.y *= sc_r;
        const v2f bm = wt[kt * 32];           // one coalesced fragment load

        accS = __builtin_amdgcn_wmma_f32_16x16x4_f32(
                 false, aS, false, bm, (short)0, accS, false, false);
        accL = __builtin_amdgcn_wmma_f32_16x16x4_f32(
                 false, aL, false, bm, (short)0, accL, false, false);
        accR = __builtin_amdgcn_wmma_f32_16x16x4_f32(
                 false, aR, false, bm, (short)0, accR, false, false);
    }

    const float bv = gb[col];
#pragma unroll
    for (int v = 0; v < 8; ++v) {
        const int row = m0 + v + 8 * half;
        const int nr  = row % N;
        const float dr = chain_dinv<N>(nr);                      // dinv[i]
        out[(long)row * C + col] =
            dr * (accS[v] + accL[v] + accR[v]) + hi[(long)row * C + col] + bv;
    }
}

// ---------------------------------------------------------------------------
// Head: out[b] = sum_{t,c} h[b,t,c] * lin_w[0, c*2+t] + lin_b  (h: [B,2,128])
// ---------------------------------------------------------------------------
__global__ void final_linear(const float* __restrict__ h,
                             const float* __restrict__ lw,
                             const float* __restrict__ lb,
                             float* __restrict__ out) {
    const int b = blockIdx.x;
    const int lane = threadIdx.x;
    float s = 0.f;
#pragma unroll
    for (int j = 0; j < 8; ++j) {
        const int e = lane + 32 * j;      // e = t*128 + c
        const int t = e >> 7;
        const int c = e & 127;
        s += h[(long)b * 256 + e] * lw[c * 2 + t];
    }
#pragma unroll
    for (int off = 16; off > 0; off >>= 1)
        s += __shfl_xor(s, off, 32);
    if (lane == 0) out[b] = s + lb[0];
}

// ---------------------------------------------------------------------------
// Launch helpers.
// ---------------------------------------------------------------------------
template <int CIN, int COUT>
static inline float* prep_conv_w(const float* w, float*& cur, hipStream_t s) {
    constexpr int K = 2 * CIN, KT = (K + 3) / 4, CT = COUT / 16;
    constexpr int T = CT * KT * 32;
    float* frag = cur; cur += 2 * T;
    repack_conv_w<CIN, COUT><<<(T + 255) / 256, 256, 0, s>>>(w, frag);
    return frag;
}

template <int C>
static inline float* prep_gcn_w(const float* gw, float*& cur, hipStream_t s) {
    constexpr int T = (C / 16) * (C / 4) * 32;
    float* frag = cur; cur += 2 * T;
    repack_gcn_w<C><<<(T + 255) / 256, 256, 0, s>>>(gw, frag);
    return frag;
}

template <int NOUT, int NIN, int CIN, int COUT, bool RELU, bool FIRST, int L>
static inline void launch_conv(const float* in, const float* wfrag, const float* b,
                               float* out, hipStream_t s) {
    const int M = 512 * NOUT;                      // multiple of 64
    dim3 grid(M / 64, COUT / 16);
    conv_wmma<NOUT, NIN, CIN, COUT, RELU, FIRST, L><<<grid, 128, 0, s>>>(in, wfrag, b, out);
}

template <int N, int C>
static inline void launch_gcn(const float* hi, const float* wfrag, const float* gb,
                              float* out, hipStream_t s) {
    const int M = 512 * N;
    dim3 grid(M / 64, C / 16);
    gcn_wmma<N, C><<<grid, 128, 0, s>>>(hi, wfrag, gb, out);
}

// ---------------------------------------------------------------------------
extern "C" void kernel_launch(void* const* d_in, const int* in_sizes, int n_in,
                              void* d_out, int out_size, void* d_ws, size_t ws_size,
                              hipStream_t stream) {
    // Input order: x | conv_w[11] | conv_b[11] | gnn_w[10] | gnn_b[10] |
    //              ew[10] | lin_w | lin_b | ei[10]
    const float* x = (const float*)d_in[0];
    const float* cw[11]; const float* cb[11];
    const float* gwp[10]; const float* gbp[10];
    for (int i = 0; i < 11; ++i) cw[i]  = (const float*)d_in[1 + i];
    for (int i = 0; i < 11; ++i) cb[i]  = (const float*)d_in[12 + i];
    for (int i = 0; i < 10; ++i) gwp[i] = (const float*)d_in[23 + i];
    for (int i = 0; i < 10; ++i) gbp[i] = (const float*)d_in[33 + i];
    const float* linw = (const float*)d_in[53];
    const float* linb = (const float*)d_in[54];

    // Workspace: ping-pong activation buffers (layer-0 sized), head buffer,
    // then fragment-major weight staging (~0.5 MB total).
    const long S0 = 512L * 3000 * 16;              // 24.576M floats (98.3 MB)
    float* A  = (float*)d_ws;                      // h_i after each layer
    float* Bf = A + S0;                            // conv output hi
    float* Cf = Bf + S0;                           // head conv out [B*2, 128]
    float* cur = Cf + 512 * 256;                   // weight-fragment arena

    // Repack all weights first (tiny kernels, same stream => ordered).
    float* cf0  = prep_conv_w<3, 16>(cw[0], cur, stream);
    float* cf1  = prep_conv_w<16, 16>(cw[1], cur, stream);
    float* cf2  = prep_conv_w<16, 16>(cw[2], cur, stream);
    float* cf3  = prep_conv_w<16, 32>(cw[3], cur, stream);
    float* cf4  = prep_conv_w<32, 32>(cw[4], cur, stream);
    float* cf5  = prep_conv_w<32, 32>(cw[5], cur, stream);
    float* cf6  = prep_conv_w<32, 64>(cw[6], cur, stream);
    float* cf7  = prep_conv_w<64, 64>(cw[7], cur, stream);
    float* cf8  = prep_conv_w<64, 64>(cw[8], cur, stream);
    float* cf9  = prep_conv_w<64, 128>(cw[9], cur, stream);
    float* cf10 = prep_conv_w<128, 128>(cw[10], cur, stream);
    float* gf0 = prep_gcn_w<16>(gwp[0], cur, stream);
    float* gf1 = prep_gcn_w<16>(gwp[1], cur, stream);
    float* gf2 = prep_gcn_w<16>(gwp[2], cur, stream);
    float* gf3 = prep_gcn_w<32>(gwp[3], cur, stream);
    float* gf4 = prep_gcn_w<32>(gwp[4], cur, stream);
    float* gf5 = prep_gcn_w<32>(gwp[5], cur, stream);
    float* gf6 = prep_gcn_w<64>(gwp[6], cur, stream);
    float* gf7 = prep_gcn_w<64>(gwp[7], cur, stream);
    float* gf8 = prep_gcn_w<64>(gwp[8], cur, stream);
    float* gf9 = prep_gcn_w<128>(gwp[9], cur, stream);

    // Layer 0 (x is [B,3,6000])
    launch_conv<3000, 6000, 3, 16, false, true, 6000>(x, cf0, cb[0], Bf, stream);
    launch_gcn<3000, 16>(Bf, gf0, gbp[0], A, stream);
    // Layers 1..9 (node-major activations, relu on conv input)
    launch_conv<1500, 3000, 16, 16, true, false, 0>(A, cf1, cb[1], Bf, stream);
    launch_gcn<1500, 16>(Bf, gf1, gbp[1], A, stream);
    launch_conv<750, 1500, 16, 16, true, false, 0>(A, cf2, cb[2], Bf, stream);
    launch_gcn<750, 16>(Bf, gf2, gbp[2], A, stream);
    launch_conv<375, 750, 16, 32, true, false, 0>(A, cf3, cb[3], Bf, stream);
    launch_gcn<375, 32>(Bf, gf3, gbp[3], A, stream);
    launch_conv<187, 375, 32, 32, true, false, 0>(A, cf4, cb[4], Bf, stream);
    launch_gcn<187, 32>(Bf, gf4, gbp[4], A, stream);
    launch_conv<93, 187, 32, 32, true, false, 0>(A, cf5, cb[5], Bf, stream);
    launch_gcn<93, 32>(Bf, gf5, gbp[5], A, stream);
    launch_conv<46, 93, 32, 64, true, false, 0>(A, cf6, cb[6], Bf, stream);
    launch_gcn<46, 64>(Bf, gf6, gbp[6], A, stream);
    launch_conv<23, 46, 64, 64, true, false, 0>(A, cf7, cb[7], Bf, stream);
    launch_gcn<23, 64>(Bf, gf7, gbp[7], A, stream);
    launch_conv<11, 23, 64, 64, true, false, 0>(A, cf8, cb[8], Bf, stream);
    launch_gcn<11, 64>(Bf, gf8, gbp[8], A, stream);
    launch_conv<5, 11, 64, 128, true, false, 0>(A, cf9, cb[9], Bf, stream);
    launch_gcn<5, 128>(Bf, gf9, gbp[9], A, stream);

    // Final conv: Nin=5 -> Nout=2, Cin=128, Cout=128, relu on input.
    launch_conv<2, 5, 128, 128, true, false, 0>(A, cf10, cb[10], Cf, stream);

    final_linear<<<512, 32, 0, stream>>>(Cf, linw, linb, (float*)d_out);
}